// DependencyNeuralModel_87359634800915
// MI455X (gfx1250) — compile-verified
//
#include <hip/hip_runtime.h>
#include <math.h>

// ---------------------------------------------------------------------------
// Sizes from the reference
// ---------------------------------------------------------------------------
#define SEQ    256
#define WE     100
#define TE     20
#define IN_DIM (WE + TE)       // 120
#define H      400
#define G4     (4 * H)         // 1600
#define M_DIM  200
#define D2H    (2 * H)         // 800
#define NPROJ  12
#define NFLAT  (NPROJ * M_DIM) // 2400
#define N_ARC  65536
#define N_SIB  200000
#define N_GP   200000
#define N_GSIB 300000

typedef __attribute__((ext_vector_type(2))) float v2f;
typedef __attribute__((ext_vector_type(8))) float v8f;

__device__ __forceinline__ float sigmoidf_(float x) {
    return 1.0f / (1.0f + __expf(-x));
}

__device__ __forceinline__ v8f wmma_f32(v2f a, v2f b, v8f c) {
    return __builtin_amdgcn_wmma_f32_16x16x4_f32(false, a, false, b, (short)0, c,
                                                 false, false);
}

// ---------------------------------------------------------------------------
// 1) Embedding gather: x[s][0:100] = word_emb[words[s]], x[s][100:120] = tag_emb[tags[s]]
// ---------------------------------------------------------------------------
__global__ void embed_kernel(const int* __restrict__ words, const int* __restrict__ tags,
                             const float* __restrict__ wemb, const float* __restrict__ temb,
                             float* __restrict__ x) {
    const int s = blockIdx.x;
    const int t = threadIdx.x;
    if (t < WE)            x[s * IN_DIM + t] = wemb[words[s] * WE + t];
    else if (t < IN_DIM)   x[s * IN_DIM + t] = temb[tags[s] * TE + (t - WE)];
}

// ---------------------------------------------------------------------------
// 2) Input GEMM via V_WMMA_F32_16X16X4_F32, 2x2 register-blocked:
//    each wave computes a 32x32 tile of z[256][1600] -> 1 load per WMMA.
//    z_f[t] = x[t] @ Wih_f^T + b_f ;  z_b[t] = x[SEQ-1-t] @ Wih_b^T + b_b
//    Grid: (8, 50, 2), block: 32 threads (EXEC all ones, required for WMMA).
// ---------------------------------------------------------------------------
__global__ void igemm_kernel(const float* __restrict__ x,
                             const float* __restrict__ Wih_f, const float* __restrict__ b_f,
                             const float* __restrict__ Wih_b, const float* __restrict__ b_b,
                             float* __restrict__ z_f, float* __restrict__ z_b) {
    const int lane = threadIdx.x & 31;
    const int m0 = blockIdx.x * 32;
    const int n0 = blockIdx.y * 32;
    const int dir = blockIdx.z;

    const float* W    = dir ? Wih_b : Wih_f;
    const float* bias = dir ? b_b : b_f;
    float*       z    = dir ? z_b : z_f;

    const int mrow0 = m0 + (lane & 15);                 // A rows, tile 0
    const int mrow1 = mrow0 + 16;                       // A rows, tile 1
    const int arow0 = dir ? (SEQ - 1 - mrow0) : mrow0;  // reversed seq for backward dir
    const int arow1 = dir ? (SEQ - 1 - mrow1) : mrow1;
    const int kh    = (lane >> 4) * 2;                  // K-pair select per A/B layout
    const int ncol0 = n0 + (lane & 15);                 // B columns, tile 0
    const int ncol1 = ncol0 + 16;                       // B columns, tile 1

    v8f c00 = {}, c01 = {}, c10 = {}, c11 = {};
    for (int k = 0; k < IN_DIM; k += 4) {
        v2f a0 = *(const v2f*)(x + arow0 * IN_DIM + k + kh);
        v2f a1 = *(const v2f*)(x + arow1 * IN_DIM + k + kh);
        v2f b0 = *(const v2f*)(W + ncol0 * IN_DIM + k + kh);  // B[k..][n] = Wih[n][k..]
        v2f b1 = *(const v2f*)(W + ncol1 * IN_DIM + k + kh);
        c00 = wmma_f32(a0, b0, c00);
        c01 = wmma_f32(a0, b1, c01);
        c10 = wmma_f32(a1, b0, c10);
        c11 = wmma_f32(a1, b1, c11);
    }
    // C/D layout: vgpr r -> M = tile_m0 + r + 8*(lane>>4), N = tile_n0 + (lane&15)
    const float bn0 = bias[ncol0];
    const float bn1 = bias[ncol1];
    const int mb = m0 + (lane >> 4) * 8;
#pragma unroll
    for (int r = 0; r < 8; ++r) {
        z[(mb + r) * G4 + ncol0]      = c00[r] + bn0;
        z[(mb + r) * G4 + ncol1]      = c01[r] + bn1;
        z[(mb + 16 + r) * G4 + ncol0] = c10[r] + bn0;
        z[(mb + 16 + r) * G4 + ncol1] = c11[r] + bn1;
    }
}

// ---------------------------------------------------------------------------
// 3) LSTM recurrence (serial over 256 steps). One workgroup per direction.
//    h, c, z kept in LDS; Whh streamed from L2 as float4.
// ---------------------------------------------------------------------------
__global__ void lstm_kernel(const float* __restrict__ z_f, const float* __restrict__ z_b,
                            const float* __restrict__ Whh_f, const float* __restrict__ Whh_b,
                            float* __restrict__ states) {
    const int dir = blockIdx.x;
    const float* z_in = dir ? z_b : z_f;
    const float* Whh  = dir ? Whh_b : Whh_f;

    __shared__ __align__(16) float h_s[H];
    __shared__ __align__(16) float c_s[H];
    __shared__ __align__(16) float z_s[G4];

    const int tid = threadIdx.x;   // blockDim.x == 1024
    for (int u = tid; u < H; u += 1024) { h_s[u] = 0.0f; c_s[u] = 0.0f; }
    __syncthreads();

    for (int t = 0; t < SEQ; ++t) {
        // z = z_in[t] + Whh @ h        (1600 rows over 1024 threads)
        for (int row = tid; row < G4; row += 1024) {
            const float4* wr = (const float4*)(Whh + row * H);
            float acc = 0.0f;
#pragma unroll 4
            for (int k4 = 0; k4 < H / 4; ++k4) {
                float4 w  = wr[k4];
                float4 hv = *(const float4*)(h_s + 4 * k4);
                acc += w.x * hv.x + w.y * hv.y + w.z * hv.z + w.w * hv.w;
            }
            z_s[row] = z_in[t * G4 + row] + acc;
            if (t + 1 < SEQ)   // warm next step's input contribution into cache
                __builtin_prefetch(z_in + (t + 1) * G4 + row, 0, 1);
        }
        __syncthreads();
        // gates: z = [i | f | g | o]
        if (tid < H) {
            float i = sigmoidf_(z_s[tid]);
            float f = sigmoidf_(z_s[tid + H]);
            float g = tanhf(z_s[tid + 2 * H]);
            float o = sigmoidf_(z_s[tid + 3 * H]);
            float cn = f * c_s[tid] + i * g;
            float hn = o * tanhf(cn);
            c_s[tid] = cn;
            h_s[tid] = hn;
            const int spos = dir ? (SEQ - 1 - t) : t;
            states[spos * D2H + dir * H + tid] = hn;
        }
        __syncthreads();
    }
}

// ---------------------------------------------------------------------------
// 4) Fused projection GEMM via WMMA, 2x2 register-blocked:
//    proj[s][p*200+m] = states[s] . W_proj[p][m]
//    A = states [256 x 800], B^T = W_proj flat [2400 x 800], C = [256 x 2400].
//    N = 2400 = 75 exact 32-wide tiles. Grid: (8, 75), block: 32.
// ---------------------------------------------------------------------------
__global__ void proj_kernel(const float* __restrict__ states,
                            const float* __restrict__ Wp,   // [2400][800]
                            float* __restrict__ proj) {     // [256][2400]
    const int lane = threadIdx.x & 31;
    const int m0 = blockIdx.x * 32;
    const int n0 = blockIdx.y * 32;
    const int mrow0 = m0 + (lane & 15);
    const int mrow1 = mrow0 + 16;
    const int kh    = (lane >> 4) * 2;
    const int ncol0 = n0 + (lane & 15);
    const int ncol1 = ncol0 + 16;

    v8f c00 = {}, c01 = {}, c10 = {}, c11 = {};
    for (int k = 0; k < D2H; k += 4) {
        v2f a0 = *(const v2f*)(states + mrow0 * D2H + k + kh);
        v2f a1 = *(const v2f*)(states + mrow1 * D2H + k + kh);
        v2f b0 = *(const v2f*)(Wp + ncol0 * D2H + k + kh);
        v2f b1 = *(const v2f*)(Wp + ncol1 * D2H + k + kh);
        c00 = wmma_f32(a0, b0, c00);
        c01 = wmma_f32(a0, b1, c01);
        c10 = wmma_f32(a1, b0, c10);
        c11 = wmma_f32(a1, b1, c11);
    }
    const int mb = m0 + (lane >> 4) * 8;
#pragma unroll
    for (int r = 0; r < 8; ++r) {
        proj[(mb + r) * NFLAT + ncol0]      = c00[r];
        proj[(mb + r) * NFLAT + ncol1]      = c01[r];
        proj[(mb + 16 + r) * NFLAT + ncol0] = c10[r];
        proj[(mb + 16 + r) * NFLAT + ncol1] = c11[r];
    }
}

// ---------------------------------------------------------------------------
// 5) Scoring: one wave per tuple, lane-strided over M=200, wave32 butterfly
//    reduction. Gathered rows are 800 B contiguous (L2-resident tables).
//    sib index == SEQ selects the null_sib row.
// ---------------------------------------------------------------------------
__global__ void score_kernel(const float* __restrict__ proj,           // [256][2400]
                             const int* __restrict__ ih, const int* __restrict__ im,
                             const int* __restrict__ is, const int* __restrict__ ig,
                             int ph, int pm, int ps, int pg,           // column offsets p*200
                             const float* __restrict__ null_sib,
                             const float* __restrict__ wsc,            // [200]
                             float* __restrict__ out, int N) {
    const int lane   = threadIdx.x & 31;
    const int wid    = (blockIdx.x * blockDim.x + threadIdx.x) >> 5;
    const int nwaves = (gridDim.x * blockDim.x) >> 5;

    for (int tup = wid; tup < N; tup += nwaves) {
        const float* rh = proj + ih[tup] * NFLAT + ph;
        const float* rm = proj + im[tup] * NFLAT + pm;
        const float* rs = nullptr;
        const float* rg = nullptr;
        if (is) {
            const int s = is[tup];
            rs = (s >= SEQ) ? null_sib : (proj + s * NFLAT + ps);
        }
        if (ig) rg = proj + ig[tup] * NFLAT + pg;

        float acc = 0.0f;
        for (int m = lane; m < M_DIM; m += 32) {
            float v = rh[m] + rm[m];
            if (rs) v += rs[m];
            if (rg) v += rg[m];
            acc += tanhf(v) * wsc[m];
        }
#pragma unroll
        for (int off = 16; off > 0; off >>= 1)
            acc += __shfl_xor(acc, off, 32);   // wave32 reduction
        if (lane == 0) out[tup] = acc;
    }
}

// ---------------------------------------------------------------------------
// Host-side launcher
// ---------------------------------------------------------------------------
extern "C" void kernel_launch(void* const* d_in, const int* in_sizes, int n_in,
                              void* d_out, int out_size, void* d_ws, size_t ws_size,
                              hipStream_t stream) {
    const int* words      = (const int*)d_in[0];
    const int* tags       = (const int*)d_in[1];
    const int* arc_head   = (const int*)d_in[2];
    const int* arc_mod    = (const int*)d_in[3];
    const int* sib_head   = (const int*)d_in[4];
    const int* sib_mod    = (const int*)d_in[5];
    const int* sib_sib    = (const int*)d_in[6];
    const int* gp_head    = (const int*)d_in[7];
    const int* gp_mod     = (const int*)d_in[8];
    const int* gp_grand   = (const int*)d_in[9];
    const int* gsib_head  = (const int*)d_in[10];
    const int* gsib_mod   = (const int*)d_in[11];
    const int* gsib_sib   = (const int*)d_in[12];
    const int* gsib_grand = (const int*)d_in[13];
    const float* word_emb = (const float*)d_in[14];
    const float* tag_emb  = (const float*)d_in[15];
    const float* Wih_f    = (const float*)d_in[16];
    const float* Whh_f    = (const float*)d_in[17];
    const float* b_f      = (const float*)d_in[18];
    const float* Wih_b    = (const float*)d_in[19];
    const float* Whh_b    = (const float*)d_in[20];
    const float* b_b      = (const float*)d_in[21];
    const float* W_proj   = (const float*)d_in[22];   // [12][200][800] = [2400][800] flat
    const float* W_score  = (const float*)d_in[23];   // [4][200]
    const float* null_sib = (const float*)d_in[24];   // [200]

    float* out = (float*)d_out;

    // Workspace carve-up (floats)
    float* ws      = (float*)d_ws;
    float* x       = ws;                               // 256*120    = 30720
    float* z_f     = x + SEQ * IN_DIM;                 // 256*1600   = 409600
    float* z_b     = z_f + SEQ * G4;                   // 256*1600
    float* states  = z_b + SEQ * G4;                   // 256*800    = 204800
    float* projbuf = states + SEQ * D2H;               // 256*2400   = 614400
    // total ~6.7 MB

    // 1) embeddings
    embed_kernel<<<SEQ, 128, 0, stream>>>(words, tags, word_emb, tag_emb, x);

    // 2) input GEMMs (both directions), WMMA f32, 32x32 tile per wave
    igemm_kernel<<<dim3(SEQ / 32, G4 / 32, 2), 32, 0, stream>>>(
        x, Wih_f, b_f, Wih_b, b_b, z_f, z_b);

    // 3) bidirectional LSTM recurrence
    lstm_kernel<<<2, 1024, 0, stream>>>(z_f, z_b, Whh_f, Whh_b, states);

    // 4) fused 12-way projection, WMMA f32, 32x32 tile per wave
    proj_kernel<<<dim3(SEQ / 32, NFLAT / 32), 32, 0, stream>>>(states, W_proj, projbuf);

    // 5) four scoring passes; proj order: 0 a_h, 1 a_m, 2 g_g, 3 g_h, 4 g_m,
    //    5 s_h, 6 s_m, 7 s_s, 8 gs_h, 9 gs_m, 10 gs_s, 11 gs_g
    const dim3 sgrid(1024), sblk(256);
    // arc: tanh(a_h[h] + a_m[m]) . W_score[0]
    score_kernel<<<sgrid, sblk, 0, stream>>>(projbuf, arc_head, arc_mod, nullptr, nullptr,
                                             0 * M_DIM, 1 * M_DIM, 0, 0,
                                             null_sib, W_score + 0 * M_DIM,
                                             out, N_ARC);
    // sib: tanh(s_h[h] + s_m[m] + sib_tab[s]) . W_score[1]
    score_kernel<<<sgrid, sblk, 0, stream>>>(projbuf, sib_head, sib_mod, sib_sib, nullptr,
                                             5 * M_DIM, 6 * M_DIM, 7 * M_DIM, 0,
                                             null_sib, W_score + 1 * M_DIM,
                                             out + N_ARC, N_SIB);
    // gp: tanh(g_h[h] + g_m[m] + g_g[g]) . W_score[2]   (grand via the ig slot)
    score_kernel<<<sgrid, sblk, 0, stream>>>(projbuf, gp_head, gp_mod, nullptr, gp_grand,
                                             3 * M_DIM, 4 * M_DIM, 0, 2 * M_DIM,
                                             null_sib, W_score + 2 * M_DIM,
                                             out + N_ARC + N_SIB, N_GP);
    // gsib: tanh(gs_h[h] + gs_m[m] + gsib_tab[s] + gs_g[g]) . W_score[3]
    score_kernel<<<sgrid, sblk, 0, stream>>>(projbuf, gsib_head, gsib_mod, gsib_sib, gsib_grand,
                                             8 * M_DIM, 9 * M_DIM, 10 * M_DIM, 11 * M_DIM,
                                             null_sib, W_score + 3 * M_DIM,
                                             out + N_ARC + N_SIB + N_GP, N_GSIB);
}